// LinearChainCRF_46454366273573
// MI455X (gfx1250) — compile-verified
//
#include <hip/hip_runtime.h>

#define B_TOT 256
#define S_LEN 2048
#define K_DIM 64
#define GROUP_B 16   // batches per workgroup (WMMA M dimension)

typedef __attribute__((ext_vector_type(2))) float v2f;
typedef __attribute__((ext_vector_type(8))) float v8f;

// Raw hardware transcendentals (v_log_f32 = log2, v_exp_f32 = exp2).
// Arguments in this kernel are never subnormal, so skip libm's fixup code.
__device__ __forceinline__ float fast_log(float x) {
    return __builtin_amdgcn_logf(x) * 0.69314718055994531f;
}
__device__ __forceinline__ float fast_exp(float x) {
    return __builtin_amdgcn_exp2f(x * 1.44269504088896341f);
}

// ---------------------------------------------------------------------------
// Forward (log-partition) kernel: serial scan over S, exp-domain recurrence
// with per-batch max shift, K x K transition handled by V_WMMA_F32_16X16X4_F32.
// 16 workgroups x 128 threads (4 waves). Workgroup g owns batches [16g,16g+16).
// Wave w owns output columns j in [16w, 16w+16).
// ---------------------------------------------------------------------------
__global__ __launch_bounds__(128) void crf_forward_kernel(
    const float* __restrict__ scores,      // [B, S, K]
    const float* __restrict__ transition,  // [K, K]
    const float* __restrict__ source,      // [K]
    const float* __restrict__ sink,        // [K]
    float* __restrict__ logZ)              // [B]
{
    // eaT[i][b]: exp(alpha[b][i] - m[b]), stride 17 -> conflict-free column gather
    __shared__ float eaT[K_DIM * 17];
    // anew[b][j]: alpha_t, stride 65 padding
    __shared__ float anew[GROUP_B * 65];
    __shared__ float m_lds[GROUP_B];

    const int tid  = threadIdx.x;
    const int lane = tid & 31;
    const int wv   = tid >> 5;       // wave id = j-tile
    const int hi   = lane >> 4;      // half-wave (K split of WMMA operands)
    const int lj   = lane & 15;
    const int jcol = wv * 16 + lj;   // this lane's column 0..63
    const int b0   = blockIdx.x * GROUP_B;

    // B-matrix fragments of E = exp(transition): constant across all 2048 steps.
    // 16x16x4 B layout: lanes 0-15 hold N, VGPR0=K(2*hi), VGPR1=K(2*hi)+1.
    v2f bk[16];
#pragma unroll
    for (int c = 0; c < 16; ++c) {
        int i0 = 4 * c + 2 * hi;
        bk[c].x = fast_exp(transition[i0 * K_DIM + jcol]);
        bk[c].y = fast_exp(transition[(i0 + 1) * K_DIM + jcol]);
    }

    // (b,q) thread layout for the renormalize phase
    const int pb  = tid >> 3;   // batch 0..15
    const int pq  = tid & 7;
    const int pi0 = pq * 8;     // i range [pi0, pi0+8)

    // ---- alpha_0 = source + scores[:,0,:]  ->  m, eaT ----
    {
        float x[8];
        float m = -3.4e38f;
#pragma unroll
        for (int r = 0; r < 8; ++r) {
            int i = pi0 + r;
            x[r] = source[i] + scores[(size_t)(b0 + pb) * S_LEN * K_DIM + i];
            m = fmaxf(m, x[r]);
        }
        m = fmaxf(m, __shfl_xor(m, 1, 8));
        m = fmaxf(m, __shfl_xor(m, 2, 8));
        m = fmaxf(m, __shfl_xor(m, 4, 8));
#pragma unroll
        for (int r = 0; r < 8; ++r)
            eaT[(pi0 + r) * 17 + pb] = fast_exp(x[r] - m);
        if (pq == 0) m_lds[pb] = m;
    }
    __syncthreads();

    // Prefetch scores for t=1 (C/D layout: row M = r + 8*hi, col = jcol)
    float sc[8];
#pragma unroll
    for (int r = 0; r < 8; ++r)
        sc[r] = scores[(size_t)(b0 + r + 8 * hi) * S_LEN * K_DIM
                       + (size_t)1 * K_DIM + jcol];

    for (int t = 1; t < S_LEN; ++t) {
        // Software pipeline: issue t+1 loads (clamped, branchless -> same
        // scheduling region as the rest of the step body).
        const int tn = (t + 1 < S_LEN) ? (t + 1) : t;
        float scn[8];
#pragma unroll
        for (int r = 0; r < 8; ++r)
            scn[r] = scores[(size_t)(b0 + r + 8 * hi) * S_LEN * K_DIM
                            + (size_t)tn * K_DIM + jcol];

        // previous max shift per batch row
        float mp[8];
#pragma unroll
        for (int r = 0; r < 8; ++r) mp[r] = m_lds[r + 8 * hi];

        // A-fragments (A layout: lane M=lj, VGPR0=K(2hi), VGPR1=K(2hi)+1)
        v2f a[16];
#pragma unroll
        for (int c = 0; c < 16; ++c) {
            int i0 = 4 * c + 2 * hi;
            a[c].x = eaT[i0 * 17 + lj];
            a[c].y = eaT[(i0 + 1) * 17 + lj];
        }

        // s[b][j] = sum_i exp(alpha[b][i]-m[b]) * E[i][j]
        // 4 independent accumulator chains (dep distance 4) so the XDL pipe
        // issues at full rate instead of stalling on C<-D latency each WMMA.
        v8f acc[4];
#pragma unroll
        for (int k = 0; k < 4; ++k)
            acc[k] = (v8f){0.f, 0.f, 0.f, 0.f, 0.f, 0.f, 0.f, 0.f};
#pragma unroll
        for (int c = 0; c < 16; ++c)
            acc[c & 3] = __builtin_amdgcn_wmma_f32_16x16x4_f32(
                false, a[c], false, bk[c], (short)0, acc[c & 3], false, false);

        v8f s01 = acc[0] + acc[1];
        v8f s23 = acc[2] + acc[3];
        v8f ssum = s01 + s23;

        // alpha_new[b][j] = m[b] + log(s) + scores[b,t,j]
        float av[8];
#pragma unroll
        for (int r = 0; r < 8; ++r)
            av[r] = mp[r] + fast_log(ssum[r]) + sc[r];

        // Pin the pipeline shape for this step:
        //   prefetch first, then ALL ds reads (one wait), then the 16 WMMAs
        //   back-to-back, then the 8 v_log_f32.
        __builtin_amdgcn_sched_group_barrier(0x020, 8, 0);   // VMEM read: prefetch
        __builtin_amdgcn_sched_group_barrier(0x100, 40, 0);  // DS read: A frags + m
        __builtin_amdgcn_sched_group_barrier(0x008, 16, 0);  // MFMA/WMMA chain
        __builtin_amdgcn_sched_group_barrier(0x400, 8, 0);   // TRANS: v_log_f32

#pragma unroll
        for (int r = 0; r < 8; ++r)
            anew[(r + 8 * hi) * 65 + jcol] = av[r];
        __syncthreads();

        // renormalize: new m[b], eaT for next step
        {
            float x[8];
            float m = -3.4e38f;
#pragma unroll
            for (int r = 0; r < 8; ++r) {
                x[r] = anew[pb * 65 + pi0 + r];
                m = fmaxf(m, x[r]);
            }
            m = fmaxf(m, __shfl_xor(m, 1, 8));
            m = fmaxf(m, __shfl_xor(m, 2, 8));
            m = fmaxf(m, __shfl_xor(m, 4, 8));
#pragma unroll
            for (int r = 0; r < 8; ++r)
                eaT[(pi0 + r) * 17 + pb] = fast_exp(x[r] - m);
            if (pq == 0) m_lds[pb] = m;
        }
        __syncthreads();

#pragma unroll
        for (int r = 0; r < 8; ++r) sc[r] = scn[r];
    }

    // logZ[b] = m[b] + log( sum_i eaT[i][b] * exp(sink[i]) )
    {
        float p = 0.f;
#pragma unroll
        for (int r = 0; r < 8; ++r)
            p += eaT[(pi0 + r) * 17 + pb] * fast_exp(sink[pi0 + r]);
        p += __shfl_xor(p, 1, 8);
        p += __shfl_xor(p, 2, 8);
        p += __shfl_xor(p, 4, 8);
        if (pq == 0)
            logZ[b0 + pb] = m_lds[pb] + fast_log(p);
    }
}

// ---------------------------------------------------------------------------
// Gold-path score: one block per batch, fixed-order tree reduction.
// ---------------------------------------------------------------------------
__global__ __launch_bounds__(256) void crf_gold_kernel(
    const float* __restrict__ scores,
    const int* __restrict__ states,
    const float* __restrict__ transition,
    const float* __restrict__ source,
    const float* __restrict__ sink,
    float* __restrict__ gold)
{
    __shared__ float red[256];
    const int b = blockIdx.x, tid = threadIdx.x;
    const int* st = states + (size_t)b * S_LEN;
    const float* sb = scores + (size_t)b * S_LEN * K_DIM;

    float acc = 0.f;
    for (int t = tid; t < S_LEN; t += 256) {
        int s1 = st[t];
        acc += sb[(size_t)t * K_DIM + s1];
        if (t > 0) acc += transition[st[t - 1] * K_DIM + s1];
    }
    red[tid] = acc;
    __syncthreads();
    for (int off = 128; off > 0; off >>= 1) {
        if (tid < off) red[tid] += red[tid + off];
        __syncthreads();
    }
    if (tid == 0)
        gold[b] = red[0] + source[st[0]] + sink[st[S_LEN - 1]];
}

// ---------------------------------------------------------------------------
// Final: mean over batch of (logZ - gold)
// ---------------------------------------------------------------------------
__global__ __launch_bounds__(256) void crf_loss_kernel(
    const float* __restrict__ logZ,
    const float* __restrict__ gold,
    float* __restrict__ out)
{
    __shared__ float red[256];
    const int tid = threadIdx.x;
    red[tid] = logZ[tid] - gold[tid];
    __syncthreads();
    for (int off = 128; off > 0; off >>= 1) {
        if (tid < off) red[tid] += red[tid + off];
        __syncthreads();
    }
    if (tid == 0) out[0] = red[0] / (float)B_TOT;
}

extern "C" void kernel_launch(void* const* d_in, const int* in_sizes, int n_in,
                              void* d_out, int out_size, void* d_ws, size_t ws_size,
                              hipStream_t stream) {
    (void)in_sizes; (void)n_in; (void)out_size; (void)ws_size;
    const float* scores     = (const float*)d_in[0];
    const int*   states     = (const int*)d_in[1];
    const float* transition = (const float*)d_in[2];
    const float* source     = (const float*)d_in[3];
    const float* sink       = (const float*)d_in[4];

    float* logZ = (float*)d_ws;          // 256 floats
    float* gold = logZ + B_TOT;          // 256 floats

    crf_gold_kernel<<<B_TOT, 256, 0, stream>>>(scores, states, transition,
                                               source, sink, gold);
    crf_forward_kernel<<<B_TOT / GROUP_B, 128, 0, stream>>>(scores, transition,
                                                            source, sink, logZ);
    crf_loss_kernel<<<1, 256, 0, stream>>>(logZ, gold, (float*)d_out);
}